// aten_sparse_mm_sxs_80616536146653
// MI455X (gfx1250) — compile-verified
//
#include <hip/hip_runtime.h>
#include <hip/hip_bf16.h>

typedef __attribute__((ext_vector_type(2))) float v2f;
typedef __attribute__((ext_vector_type(8))) float v8f;

static constexpr int    MDIM  = 8192;
static constexpr int    KDIM  = 8192;
static constexpr int    NDIM  = 2048;
static constexpr int    NNZA  = 131072;
static constexpr int    NNZB  = 131072;
static constexpr int    NBINS = MDIM / 16;    // 512 row-tiles of 16 rows
static constexpr int    NGRPS = NDIM / 64;    // 32 col-groups of 64 columns
static constexpr int    WAVES_PER_BLOCK = 8;  // 8 waves * 32 lanes = 256 threads
static constexpr int    ROW_BYTES = NDIM * 4; // 8192 B per dense-B row (== k<<13)

// ---- workspace layout (bytes) --------------------------------------------
static constexpr size_t WS_BDENSE  = 0;                           // float[KDIM*NDIM] = 64 MB
static constexpr size_t WS_COUNTS  = (size_t)KDIM * NDIM * 4;     // int[NBINS]
static constexpr size_t WS_OFFSETS = WS_COUNTS  + 2048;           // int[NBINS+1]
static constexpr size_t WS_CURSORS = WS_OFFSETS + 4096;           // int[NBINS]
static constexpr size_t WS_ENTRIES = WS_CURSORS + 2048;           // uint2[NNZA + 7*NBINS], 16B-aligned

// ---- phase 1: zero dense B + bin counters --------------------------------
__global__ void zero_ws_kernel(float4* __restrict__ bdense4, int* __restrict__ counts) {
    int tid = blockIdx.x * blockDim.x + threadIdx.x;    // exactly KDIM*NDIM/4 threads
    bdense4[tid] = make_float4(0.0f, 0.0f, 0.0f, 0.0f);
    if (tid < NBINS) counts[tid] = 0;
}

// ---- phase 2: densify B via scatter-add (duplicate COO entries sum) ------
__global__ void scatter_b_kernel(const int* __restrict__ idx2,
                                 const float* __restrict__ val2,
                                 float* __restrict__ bdense) {
    int tid = blockIdx.x * blockDim.x + threadIdx.x;
    if (tid >= NNZB) return;
    int k = idx2[tid];          // row id in [0, K)
    int n = idx2[NNZB + tid];   // col id in [0, N)
    atomicAdd(bdense + (size_t)k * NDIM + n, val2[tid]);
}

// ---- phase 3: histogram A nnz by 16-row output tile ----------------------
__global__ void count_a_kernel(const int* __restrict__ idx1, int* __restrict__ counts) {
    int tid = blockIdx.x * blockDim.x + threadIdx.x;
    if (tid >= NNZA) return;
    atomicAdd(counts + (idx1[tid] >> 4), 1);
}

// ---- phase 4: exclusive scan over bins PADDED to multiples of 8 ----------
// Padding guarantees: no tail handling in the WMMA loop, 16B-aligned pair
// loads, uniform trip counts (EXEC stays all-ones, as WMMA requires).
__global__ void scan_kernel(const int* __restrict__ counts,
                            int* __restrict__ offsets,
                            int* __restrict__ cursors) {
    __shared__ int tmp[NBINS];
    int t = threadIdx.x;
    int x = (counts[t] + 7) & ~7;   // padded bin length
    tmp[t] = x;
    __syncthreads();
    #pragma unroll
    for (int d = 1; d < NBINS; d <<= 1) {
        int y = (t >= d) ? tmp[t - d] : 0;
        __syncthreads();
        tmp[t] += y;
        __syncthreads();
    }
    int incl = tmp[t];
    offsets[t] = incl - x;
    cursors[t] = incl - x;
    if (t == NBINS - 1) offsets[NBINS] = incl;   // padded total
}

// ---- phase 5a: scatter A entries into their row-tile bin -----------------
// entry = { (k << 13) | (row & 15) , f32 bits of value }
// k<<13 is the BYTE offset of dense-B row k (N*4 = 8192 = 1<<13), so the
// WMMA loop decodes the load offset with a single AND.
__global__ void scatter_a_kernel(const int* __restrict__ idx1,
                                 const float* __restrict__ val1,
                                 int* __restrict__ cursors,
                                 uint2* __restrict__ entries) {
    int tid = blockIdx.x * blockDim.x + threadIdx.x;
    if (tid >= NNZA) return;
    int r = idx1[tid];
    int k = idx1[NNZA + tid];
    float v = val1[tid];
    int pos = atomicAdd(cursors + (r >> 4), 1);
    entries[pos] = make_uint2((unsigned)((k << 13) | (r & 15)), __float_as_uint(v));
}

// ---- phase 5b: fill the per-bin padding slots with zero entries ----------
// meta=0, v=0 -> A-fragment element is 0 -> contributes nothing; its B load
// hits row 0 harmlessly.
__global__ void pad_kernel(const int* __restrict__ counts,
                           const int* __restrict__ offsets,
                           uint2* __restrict__ entries) {
    int tid = blockIdx.x * blockDim.x + threadIdx.x;
    if (tid >= NBINS * 7) return;
    int bin = tid / 7, j = tid % 7;
    int pos = offsets[bin] + counts[bin] + j;
    if (pos < offsets[bin + 1]) entries[pos] = make_uint2(0u, 0u);
}

// ---- phase 6: gather-GEMM with V_WMMA_F32_16X16X4_F32 --------------------
// One wave owns a 16x64 output tile: four 16x16 accumulators sharing one A
// fragment per K=4 chunk. Bins are padded to multiples of 8, so each
// iteration processes 8 entries with 8 WMMAs, 2 b128 entry loads, and 16
// immediate-offset b32 B loads -- completely branch-free, EXEC all-ones.
//
// A lane-masking note: a.x is built as v * (row==sub ? 1.0f : 0.0f) rather
// than a select of the loaded value; this keeps the entry load's value word
// unconditionally used, so the compiler cannot sink it under a lane
// predicate (which previously cost an EXEC save/branch/restore per
// iteration). For finite inputs v*1==v and v*0==0, so math is unchanged.
//
// A layout (ISA 16x4 f32): lanes 0-15 -> M=lane, {K0,K1}; lanes 16-31 ->
// M=lane-16, {K2,K3}. B mirrors with N striped across lanes. C/D: VGPR r ->
// M = r + 8*(lane>=16), N = lane&15.
__global__ void __launch_bounds__(256)
spmm_wmma_kernel(const char* __restrict__ bdense_bytes,
                 const int* __restrict__ offsets,
                 const uint4* __restrict__ entries4,
                 float* __restrict__ out) {
    const int tile_m = blockIdx.x;
    const int wave   = threadIdx.x >> 5;
    const int lane   = threadIdx.x & 31;
    const int grp    = blockIdx.y * WAVES_PER_BLOCK + wave;  // 64-column group

    const int bin_start = offsets[tile_m];        // multiple of 8
    const int bin_end   = offsets[tile_m + 1];

    const int      half = lane >> 4;              // 0: K entries {0,1} | 1: {2,3}
    const unsigned sub  = (unsigned)(lane & 15);  // A: local row m | B: local col n
    const unsigned colb = (unsigned)((grp * 64 + (int)sub) * 4);  // byte col offset

    v8f c0 = {}, c1 = {}, c2 = {}, c3 = {};

    auto do_chunk = [&](const uint4 e) {
        unsigned addrA = (e.x & ~8191u) + colb;   // B-row byte address, lane column
        unsigned addrB = (e.z & ~8191u) + colb;
        float m0 = ((e.x & 15u) == sub) ? 1.0f : 0.0f;
        float m1 = ((e.z & 15u) == sub) ? 1.0f : 0.0f;
        v2f a;
        a.x = __uint_as_float(e.y) * m0;
        a.y = __uint_as_float(e.w) * m1;
        v2f b0, b1, b2, b3;
        b0.x = *(const float*)(bdense_bytes + addrA);
        b0.y = *(const float*)(bdense_bytes + addrB);
        b1.x = *(const float*)(bdense_bytes + addrA + 64);
        b1.y = *(const float*)(bdense_bytes + addrB + 64);
        b2.x = *(const float*)(bdense_bytes + addrA + 128);
        b2.y = *(const float*)(bdense_bytes + addrB + 128);
        b3.x = *(const float*)(bdense_bytes + addrA + 192);
        b3.y = *(const float*)(bdense_bytes + addrB + 192);
        c0 = __builtin_amdgcn_wmma_f32_16x16x4_f32(false, a, false, b0, (short)0, c0, false, false);
        c1 = __builtin_amdgcn_wmma_f32_16x16x4_f32(false, a, false, b1, (short)0, c1, false, false);
        c2 = __builtin_amdgcn_wmma_f32_16x16x4_f32(false, a, false, b2, (short)0, c2, false, false);
        c3 = __builtin_amdgcn_wmma_f32_16x16x4_f32(false, a, false, b3, (short)0, c3, false, false);
    };

    int idx = (bin_start >> 1) + half;            // uint4 index of this lane's pair
    const int iters = (bin_end - bin_start) >> 3; // 8 entries / iteration
    for (int it = 0; it < iters; ++it, idx += 4) {
        uint4 p = entries4[idx];          // chunk 0: {meta,val} x 2
        uint4 q = entries4[idx + 2];      // chunk 1
        do_chunk(p);
        do_chunk(q);
    }

    // Store 16x64 tile: c0..c3 -> cols [0,16),[16,32),[32,48),[48,64).
    char* outp = (char*)out +
        ((size_t)(tile_m * 16 + half * 8) * NDIM + (size_t)(grp * 64 + (int)sub)) * 4;
    #pragma unroll
    for (int r = 0; r < 8; ++r) {
        char* rowp = outp + (size_t)r * ROW_BYTES;
        *(float*)(rowp)       = c0[r];
        *(float*)(rowp + 64)  = c1[r];
        *(float*)(rowp + 128) = c2[r];
        *(float*)(rowp + 192) = c3[r];
    }
}

extern "C" void kernel_launch(void* const* d_in, const int* in_sizes, int n_in,
                              void* d_out, int out_size, void* d_ws, size_t ws_size,
                              hipStream_t stream) {
    const int*   idx1 = (const int*)d_in[0];    // [2, NNZA] : rows then cols
    const float* val1 = (const float*)d_in[1];  // [NNZA]
    const int*   idx2 = (const int*)d_in[2];    // [2, NNZB] : rows then cols
    const float* val2 = (const float*)d_in[3];  // [NNZB]
    float* out = (float*)d_out;                 // [MDIM * NDIM] fp32

    char* ws = (char*)d_ws;
    float* bdense  = (float*)(ws + WS_BDENSE);
    int*   counts  = (int*)(ws + WS_COUNTS);
    int*   offsets = (int*)(ws + WS_OFFSETS);
    int*   cursors = (int*)(ws + WS_CURSORS);
    uint2* entries = (uint2*)(ws + WS_ENTRIES);

    // 1) zero dense B (64 MB) + bin counters
    zero_ws_kernel<<<(KDIM * NDIM / 4) / 256, 256, 0, stream>>>((float4*)bdense, counts);
    // 2) densify B
    scatter_b_kernel<<<(NNZB + 255) / 256, 256, 0, stream>>>(idx2, val2, bdense);
    // 3) histogram A by row-tile
    count_a_kernel<<<(NNZA + 255) / 256, 256, 0, stream>>>(idx1, counts);
    // 4) exclusive scan over padded bin lengths
    scan_kernel<<<1, NBINS, 0, stream>>>(counts, offsets, cursors);
    // 5) bin the A entries, then zero-fill the padding slots
    scatter_a_kernel<<<(NNZA + 255) / 256, 256, 0, stream>>>(idx1, val1, cursors, entries);
    pad_kernel<<<(NBINS * 7 + 255) / 256, 256, 0, stream>>>(counts, offsets, entries);
    // 6) WMMA gather-GEMM: 512 row-tiles x (32/8) col-groups, 8 waves/block
    dim3 grid(NBINS, NGRPS / WAVES_PER_BLOCK);
    spmm_wmma_kernel<<<grid, 32 * WAVES_PER_BLOCK, 0, stream>>>(
        (const char*)bdense, offsets, (const uint4*)entries, out);
}